// GMULayer_57655640982261
// MI455X (gfx1250) — compile-verified
//
#include <hip/hip_runtime.h>
#include <math.h>

// Problem constants (match reference)
#define BDIM 256
#define ODIM 512
#define IDIM 512
#define SDIM 32          // NUM_SLICES * DEGREE = 8*4
#define EPS  1e-4f

// padded LDS row strides (odd -> bank-conflict-free column access on 64 banks)
#define TL_STRIDE 513    // t-tile rows
#define XC_STRIDE 33     // X-chunk rows

typedef __attribute__((ext_vector_type(2))) float v2f;
typedef __attribute__((ext_vector_type(8))) float v8f;

// ---------------------------------------------------------------------------
// Kernel 1: t = y + eps*noise
// ---------------------------------------------------------------------------
__global__ void prep_t_kernel(const float* __restrict__ y,
                              const float* __restrict__ noise,
                              float* __restrict__ t, int n) {
    int i = blockIdx.x * blockDim.x + threadIdx.x;
    if (i < n) t[i] = y[i] + EPS * noise[i];
}

// ---------------------------------------------------------------------------
// Kernel 2: per-o Gram matrix C[o] = X[o]^T X[o]  (32x32), X = [w,w^2,w^3,w^4]
// One block per o, 128 threads = 4 waves; wave w owns C tile (w>>1, w&1).
// X chunks (64 i-rows x 32 p-cols) are generated from weights into LDS, then
// consumed by v_wmma_f32_16x16x4_f32 (both A and B fragments come from LDS).
// ---------------------------------------------------------------------------
__global__ void xtx_kernel(const float* __restrict__ weights,
                           float* __restrict__ C) {
    const int o    = blockIdx.x;
    const int tid  = threadIdx.x;        // 0..127
    const int wave = tid >> 5;
    const int lane = tid & 31;
    const int mtile = wave >> 1;
    const int ntile = wave & 1;

    __shared__ float Xl[64][SDIM];       // 8 KB chunk of X (i-major)

    const int arow = mtile * 16 + (lane & 15);   // A: M index = p-row
    const int bcol = ntile * 16 + (lane & 15);   // B: N index = p-col
    const int kg   = (lane >> 4) * 2;            // K sub-offset per lane half

    v8f acc = {};

    for (int cc = 0; cc < IDIM / 64; ++cc) {
        // ---- stage: build 64 rows of X from weights (4 powers of 8 slices)
        {
            const int il   = tid >> 1;           // 0..63 local i
            const int half = tid & 1;            // 0: p 0..15, 1: p 16..31
            const float* wp = weights + ((size_t)o * IDIM + (cc * 64 + il)) * 8;
#pragma unroll
            for (int s = 0; s < 8; ++s) {
                float w = wp[s], w2 = w * w;
                if (half == 0) { Xl[il][s]      = w;      Xl[il][8 + s]  = w2;      }
                else           { Xl[il][16 + s] = w2 * w; Xl[il][24 + s] = w2 * w2; }
            }
        }
        __syncthreads();
        // ---- 16 K-steps of 4 over this chunk
#pragma unroll
        for (int kk = 0; kk < 16; ++kk) {
            const int k0 = kk * 4 + kg;
            v2f a, b;
            a.x = Xl[k0][arow];     a.y = Xl[k0 + 1][arow];
            b.x = Xl[k0][bcol];     b.y = Xl[k0 + 1][bcol];
            acc = __builtin_amdgcn_wmma_f32_16x16x4_f32(
                      false, a, false, b, (short)0, acc, false, false);
        }
        __syncthreads();
    }

    // C/D layout: VGPR r, lanes 0-15 -> M=r, lanes 16-31 -> M=r+8; N=lane&15
#pragma unroll
    for (int r = 0; r < 8; ++r) {
        const int M = mtile * 16 + r + ((lane >> 4) << 3);
        C[(size_t)o * (SDIM * SDIM) + M * SDIM + bcol] = acc[r];
    }
}

// ---------------------------------------------------------------------------
// Kernel 3: in-place batched 32x32 inverse (Gauss-Jordan, SPD Gram matrices).
// One block (single wave, 32 threads) per o; thread r owns row r of [A | I].
// ---------------------------------------------------------------------------
__global__ void invert_kernel(float* __restrict__ C) {
    const int o = blockIdx.x;
    const int r = threadIdx.x;           // 0..31
    __shared__ float A[32][65];          // 65: odd stride -> conflict-free cols

#pragma unroll 4
    for (int c = 0; c < 32; ++c) A[r][c] = C[(size_t)o * 1024 + r * 32 + c];
#pragma unroll 4
    for (int c = 0; c < 32; ++c) A[r][32 + c] = (r == c) ? 1.0f : 0.0f;
    __syncthreads();

    for (int k = 0; k < 32; ++k) {
        if (r == k) {
            float inv = 1.0f / A[k][k];
#pragma unroll 4
            for (int c = 0; c < 64; ++c) A[k][c] *= inv;
        }
        __syncthreads();
        if (r != k) {
            float f = A[r][k];
#pragma unroll 4
            for (int c = 0; c < 64; ++c) A[r][c] -= f * A[k][c];
        }
        __syncthreads();
    }

#pragma unroll 4
    for (int c = 0; c < 32; ++c) C[(size_t)o * 1024 + r * 32 + c] = A[r][32 + c];
}

// ---------------------------------------------------------------------------
// Kernel 4 (dominant GEMM, 4.3 GFLOP): Z[o,b,p] = sum_i X[o,i,p]*(t[b,i]-wb[o,i])
// Grid (O/4, B/16); block = 256 threads = 8 waves.
// Wave w -> (o_local = w>>1, ptile = w&1).
// All WMMA operands come from LDS:
//   - t-tile (16 x 512, padded stride 513) + wb (4 x 512): A = t - wb
//   - X-chunk (4 o x 64 k x 32 p, padded stride 33): B = w^deg, powers built
//     once per element at stage time, coalesced weight loads.
// 16 f32 WMMA per wave per chunk, 8 chunks.
// ---------------------------------------------------------------------------
__global__ void zgemm_kernel(const float* __restrict__ t,
                             const float* __restrict__ weights,
                             const float* __restrict__ wb,
                             float* __restrict__ Z) {
    const int tid    = threadIdx.x;      // 0..255
    const int wave   = tid >> 5;
    const int lane   = tid & 31;
    const int olocal = wave >> 1;
    const int ptile  = wave & 1;
    const int o      = blockIdx.x * 4 + olocal;
    const int btile  = blockIdx.y;

    __shared__ float tl[16 * TL_STRIDE];        // ~32.1 KB: 16 batch rows of t
    __shared__ float wbl[4 * IDIM];             //   8 KB: weight_bias, 4 o's
    __shared__ float Xc[4 * 64 * XC_STRIDE];    //  ~33 KB: X chunk for 4 o's

    // ---- one-time staging: t tile + wb (coalesced)
#pragma unroll
    for (int j = 0; j < 32; ++j) {
        int idx = tid + 256 * j;                // 0..8191
        int row = idx >> 9, col = idx & 511;
        tl[row * TL_STRIDE + col] = t[(size_t)(btile * 16 + row) * IDIM + col];
    }
#pragma unroll
    for (int j = 0; j < 8; ++j) {
        int idx = tid + 256 * j;                // 0..2047
        wbl[idx] = wb[((size_t)blockIdx.x * 4 + (idx >> 9)) * IDIM + (idx & 511)];
    }

    const int mrow = lane & 15;                 // batch row within tile
    const int kg   = (lane >> 4) * 2;           // K sub-offset per lane half
    const int p    = ptile * 16 + (lane & 15);  // design column this lane owns

    // staging role for X chunks: one (o, i) pair per thread
    const int so = tid >> 6;                    // 0..3
    const int si = tid & 63;                    // 0..63

    v8f acc = {};

    for (int ck = 0; ck < IDIM / 64; ++ck) {
        __syncthreads();                        // previous chunk fully consumed
        // ---- stage X chunk: 8 weights -> 32 powers per thread
        {
            const float* wp = weights +
                ((size_t)(blockIdx.x * 4 + so) * IDIM + (ck * 64 + si)) * 8;
            float* xrow = &Xc[(so * 64 + si) * XC_STRIDE];
#pragma unroll
            for (int s = 0; s < 8; ++s) {
                float w = wp[s], w2 = w * w;
                xrow[s]      = w;
                xrow[8 + s]  = w2;
                xrow[16 + s] = w2 * w;
                xrow[24 + s] = w2 * w2;
            }
        }
        __syncthreads();
        // ---- 16 K-steps of 4 over this chunk, all operands from LDS
#pragma unroll
        for (int kk = 0; kk < 16; ++kk) {
            const int k0  = kk * 4 + kg;        // 0..62, chunk-local
            const int col = ck * 64 + k0;       // global k
            v2f a, b;
            a.x = tl[mrow * TL_STRIDE + col]     - wbl[olocal * IDIM + col];
            a.y = tl[mrow * TL_STRIDE + col + 1] - wbl[olocal * IDIM + col + 1];
            b.x = Xc[(olocal * 64 + k0) * XC_STRIDE + p];
            b.y = Xc[(olocal * 64 + k0 + 1) * XC_STRIDE + p];
            acc = __builtin_amdgcn_wmma_f32_16x16x4_f32(
                      false, a, false, b, (short)0, acc, false, false);
        }
    }

    // C/D layout: VGPR r, lanes 0-15 -> M=r, lanes 16-31 -> M=r+8; N=lane&15
#pragma unroll
    for (int r = 0; r < 8; ++r) {
        const int brow = btile * 16 + r + ((lane >> 4) << 3);
        Z[((size_t)o * BDIM + brow) * SDIM + p] = acc[r];
    }
}

// ---------------------------------------------------------------------------
// Kernel 5: per (b,o) scalar epilogue.
//   s1 = sum r, s2 = sum r^2 (r = t - wb[o]); quad = z^T Cinv z;
//   var = (s2 - s1^2/I)/(I-1); err = (s2 - quad)/(I*var); out = sqrt(1.000001-err)
// Block per o (Cinv + wb broadcast from LDS), thread per b.
// ---------------------------------------------------------------------------
__global__ void finalize_kernel(const float* __restrict__ t,
                                const float* __restrict__ wb,
                                const float* __restrict__ Cinv,
                                const float* __restrict__ Z,
                                float* __restrict__ out) {
    const int o = blockIdx.x;
    const int b = threadIdx.x;           // 0..255

    __shared__ float Ci[SDIM * SDIM];    // 4 KB
    __shared__ float wbl[IDIM];          // 2 KB

    for (int j = threadIdx.x; j < SDIM * SDIM; j += 256)
        Ci[j] = Cinv[(size_t)o * (SDIM * SDIM) + j];
    for (int j = threadIdx.x; j < IDIM; j += 256)
        wbl[j] = wb[(size_t)o * IDIM + j];
    __syncthreads();

    float s1 = 0.0f, s2 = 0.0f;
    const float* trow = t + (size_t)b * IDIM;
#pragma unroll 8
    for (int i = 0; i < IDIM; ++i) {
        float d = trow[i] - wbl[i];
        s1 += d;
        s2 += d * d;
    }

    float z[SDIM];
    const float* zrow = Z + ((size_t)o * BDIM + b) * SDIM;
#pragma unroll
    for (int p = 0; p < SDIM; ++p) z[p] = zrow[p];

    float quad = 0.0f;
#pragma unroll 4
    for (int p = 0; p < SDIM; ++p) {
        float yp = 0.0f;
#pragma unroll
        for (int q = 0; q < SDIM; ++q) yp += Ci[p * SDIM + q] * z[q];
        quad += z[p] * yp;
    }

    const float var = (s2 - s1 * s1 * (1.0f / IDIM)) * (1.0f / (IDIM - 1));
    const float err = (s2 - quad) / (var * (float)IDIM);
    out[(size_t)b * ODIM + o] = sqrtf(1.000001f - err);
}

// ---------------------------------------------------------------------------
// Launcher. Workspace layout (bytes):
//   [0,            524288)  t   : B*I f32
//   [524288,      2621440)  C   : O*32*32 f32  (inverted in place)
//   [2621440,    19398656)  Z   : O*B*32 f32
// Total ~18.5 MB.
// ---------------------------------------------------------------------------
extern "C" void kernel_launch(void* const* d_in, const int* in_sizes, int n_in,
                              void* d_out, int out_size, void* d_ws, size_t ws_size,
                              hipStream_t stream) {
    (void)in_sizes; (void)n_in; (void)out_size; (void)ws_size;

    const float* y       = (const float*)d_in[0];   // [B,I]
    const float* noise   = (const float*)d_in[1];   // [B,I]
    const float* weights = (const float*)d_in[2];   // [O,I,8]
    const float* wbias   = (const float*)d_in[3];   // [O,I]
    // d_in[4] = train_status (unused: computation is identical here)
    float* out = (float*)d_out;                     // [B,O,1,1] flat

    char*  ws = (char*)d_ws;
    float* t  = (float*)(ws);
    float* C  = (float*)(ws + (size_t)524288);
    float* Z  = (float*)(ws + (size_t)2621440);

    prep_t_kernel<<<(BDIM * IDIM + 255) / 256, 256, 0, stream>>>(y, noise, t, BDIM * IDIM);
    xtx_kernel   <<<ODIM, 128, 0, stream>>>(weights, C);
    invert_kernel<<<ODIM, 32, 0, stream>>>(C);
    zgemm_kernel <<<dim3(ODIM / 4, BDIM / 16), 256, 0, stream>>>(t, weights, wbias, Z);
    finalize_kernel<<<ODIM, 256, 0, stream>>>(t, wbias, C, Z, out);
}